// AttentionAggregation_72086731096444
// MI455X (gfx1250) — compile-verified
//
#include <hip/hip_runtime.h>
#include <hip/hip_bf16.h>

#define SOFT_EPS 1e-16f

typedef __attribute__((ext_vector_type(2))) float v2f;
typedef __attribute__((ext_vector_type(4))) float v4f;
typedef __attribute__((ext_vector_type(8))) float v8f;

// Order-preserving monotone map float -> uint so atomicMax(uint) == float max.
__device__ __forceinline__ unsigned enc_f32(float f) {
    unsigned b = __float_as_uint(f);
    return (b & 0x80000000u) ? ~b : (b | 0x80000000u);
}
__device__ __forceinline__ float dec_f32(unsigned u) {
    unsigned b = (u & 0x80000000u) ? (u & 0x7FFFFFFFu) : ~u;
    return __uint_as_float(b);
}

// ---------------------------------------------------------------- init
__global__ void k_init(float* __restrict__ out, float* __restrict__ denom,
                       unsigned* __restrict__ segmax, int N) {
    int i = blockIdx.x * 256 + threadIdx.x;
    if (i < N * 32) out[i] = 0.0f;
    if (i < N) { denom[i] = 0.0f; segmax[i] = 0u; }  // 0u < enc(any finite float)
}

// ------------------------------------------------- pass 1: scores + seg max
__global__ void __launch_bounds__(256)
k_scores(const float* __restrict__ ax, const int* __restrict__ idx,
         const float* __restrict__ Ws, const float* __restrict__ bs,
         float* __restrict__ sc, unsigned* __restrict__ segmax, int E) {
    int e = blockIdx.x * 256 + threadIdx.x;
    if (e >= E) return;
    const v4f* ar = (const v4f*)(ax + (size_t)e * 32);
    const v4f* wr = (const v4f*)Ws;  // thread-uniform -> scalar loads
    float s = bs[0];
#pragma unroll
    for (int i = 0; i < 8; ++i) {
        v4f a = __builtin_nontemporal_load(&ar[i]);  // 205 MB single-use stream
        v4f w = wr[i];
        s += a.x * w.x + a.y * w.y + a.z * w.z + a.w * w.w;
    }
    sc[e] = s;
    atomicMax(&segmax[idx[e]], enc_f32(s));
}

// ------------------------------------------------- pass 2: exp + denominator
__global__ void __launch_bounds__(256)
k_exp(const int* __restrict__ idx, float* __restrict__ sc,
      const unsigned* __restrict__ segmax, float* __restrict__ denom, int E) {
    int e = blockIdx.x * 256 + threadIdx.x;
    if (e >= E) return;
    int n = idx[e];
    float ex = expf(sc[e] - dec_f32(segmax[n]));
    sc[e] = ex;  // in-place: score -> exp(score - max)
    atomicAdd(&denom[n], ex);
}

// --------------------------------- pass 2.5: invert denominators (per node)
__global__ void k_invdenom(float* __restrict__ denom, int N) {
    int i = blockIdx.x * 256 + threadIdx.x;
    if (i < N) denom[i] = 1.0f / (denom[i] + SOFT_EPS);  // in place
}

// ---------------------- pass 3: WMMA fp32 GEMM (x @ W_emb) + weighted scatter
// One wave32 per 64 edges (4 x 16-edge WMMA tiles reusing B registers). D=32.
__global__ void __launch_bounds__(256)
k_gemm_scatter(const float* __restrict__ x, const int* __restrict__ idx,
               const float* __restrict__ W, const float* __restrict__ bias,
               const float* __restrict__ exv, const float* __restrict__ invden,
               float* __restrict__ out, int E) {
    const int lane = threadIdx.x & 31;
    const int wave = threadIdx.x >> 5;
    const int e0 = (blockIdx.x * 8 + wave) * 64;  // 64 edges per wave
    if (e0 + 64 > E) return;                      // wave-uniform; E % 64 == 0

    const int nlo = lane & 15;   // M index (A) / N index (B,C,D) for lanes 0-15
    const int hi  = lane >> 4;   // selects K pair (A/B) / M+8 rows (C/D)
    const int k0  = hi * 2;

    // B tiles of W_emb[32][32], element B[k][n] = W[k*32+n].
    // 32-bit B 4x16 layout: VGPR0 = rows K={0|2}, VGPR1 = rows K={1|3}
    // (lanes 0-15 -> K base 0, lanes 16-31 -> K base 2), N = lane&15.
    v2f B[2][8];
#pragma unroll
    for (int t = 0; t < 2; ++t) {
#pragma unroll
        for (int kc = 0; kc < 8; ++kc) {
            int k = kc * 4 + k0;
            int n = t * 16 + nlo;
            B[t][kc].x = W[(size_t)k * 32 + n];
            B[t][kc].y = W[(size_t)(k + 1) * 32 + n];
        }
    }

    const float bias0 = bias[nlo];
    const float bias1 = bias[16 + nlo];

#pragma unroll
    for (int sub = 0; sub < 4; ++sub) {
        const int eb = e0 + sub * 16;

        // A tile: 32-bit A 16x4 layout: lanes 0-15 hold K0/K1, lanes 16-31
        // hold K2/K3, M = lane&15 -> row eb+nlo of x. Non-temporal: x is a
        // 205 MB single-use stream; keep L2 for index/ex/denom/out atomics.
        const float* xrow = x + (size_t)(eb + nlo) * 32;
        v2f A[8];
#pragma unroll
        for (int kc = 0; kc < 8; ++kc) {
            A[kc] = __builtin_nontemporal_load((const v2f*)(xrow + kc * 4 + k0));
        }

        v8f acc0 = {};
        v8f acc1 = {};
#pragma unroll
        for (int kc = 0; kc < 8; ++kc) {
            acc0 = __builtin_amdgcn_wmma_f32_16x16x4_f32(
                false, A[kc], false, B[0][kc], (short)0, acc0, false, false);
            acc1 = __builtin_amdgcn_wmma_f32_16x16x4_f32(
                false, A[kc], false, B[1][kc], (short)0, acc1, false, false);
        }

        // C/D layout: VGPR r holds row M = r + 8*hi, column N = lane&15.
#pragma unroll
        for (int r = 0; r < 8; ++r) {
            int e = eb + hi * 8 + r;
            int node = idx[e];                       // broadcast within half-wave
            float w = exv[e] * invden[node];         // softmax weight (no divide)
            size_t o = (size_t)node * 32;
            atomicAdd(&out[o + nlo],      (acc0[r] + bias0) * w);
            atomicAdd(&out[o + 16 + nlo], (acc1[r] + bias1) * w);
        }
    }
}

// ----------------------------------------------------------------- launcher
extern "C" void kernel_launch(void* const* d_in, const int* in_sizes, int n_in,
                              void* d_out, int out_size, void* d_ws, size_t ws_size,
                              hipStream_t stream) {
    const float* x     = (const float*)d_in[0];
    const float* ax    = (const float*)d_in[1];
    const int*   idx   = (const int*)d_in[2];
    // d_in[3] = size (device scalar) -- N derived from out_size instead
    const float* W_emb = (const float*)d_in[4];
    const float* b_emb = (const float*)d_in[5];
    const float* W_sc  = (const float*)d_in[6];
    const float* b_sc  = (const float*)d_in[7];
    float* out = (float*)d_out;

    const int E = in_sizes[2];
    const int N = out_size / 32;

    // Workspace layout: ex[E] | denom[N] | segmax[N]
    float*    ws_ex  = (float*)d_ws;
    float*    denom  = ws_ex + E;
    unsigned* segmax = (unsigned*)(denom + N);

    int initElems = N * 32;
    k_init<<<(initElems + 255) / 256, 256, 0, stream>>>(out, denom, segmax, N);
    k_scores<<<(E + 255) / 256, 256, 0, stream>>>(ax, idx, W_sc, b_sc, ws_ex, segmax, E);
    k_exp<<<(E + 255) / 256, 256, 0, stream>>>(idx, ws_ex, segmax, denom, E);
    k_invdenom<<<(N + 255) / 256, 256, 0, stream>>>(denom, N);

    int waves  = E / 64;              // 64 edges per wave; E % 64 == 0
    int blocks = (waves + 7) / 8;     // 8 waves (256 threads) per block
    k_gemm_scatter<<<blocks, 256, 0, stream>>>(x, idx, W_emb, b_emb, ws_ex, denom, out, E);
}